// RegionProposal_74955769249840
// MI455X (gfx1250) — compile-verified
//
#include <hip/hip_runtime.h>
#include <cstdint>

// ---------------- problem constants ----------------
#define HH     100
#define WWID   100
#define AANC   9
#define NN     (HH * WWID * AANC)   // 90000
#define KPRE   6000
#define KPOST  300
#define NWORDS 188                  // ceil(KPRE/32)
#define NH     375                  // 16-col half-words per mask row (375*16 == 6000)
#define NTILE  375                  // 16-row tiles (375*16 == 6000)
#define MIN_SZ 0.02f                // 16/800
#define IOUTHR 0.7f
#define NBUCKET 65536

typedef __attribute__((ext_vector_type(2))) float v2f;
typedef __attribute__((ext_vector_type(8))) float v8f;

// ---------------- workspace layout (bytes) ----------------
constexpr size_t OFF_BOXES = 0;                                  // N * float4
constexpr size_t OFF_KEYS  = OFF_BOXES + (size_t)NN * 16;        // N * u32
constexpr size_t OFF_HIST  = OFF_KEYS  + (size_t)NN * 4;         // 65536 * u32
constexpr size_t OFF_CUM   = OFF_HIST  + (size_t)NBUCKET * 4;    // 65536 * u32
constexpr size_t OFF_META  = OFF_CUM   + (size_t)NBUCKET * 4;    // 16 * u32
constexpr size_t OFF_CAND  = OFF_META  + 64;                     // N * uint2 (worst case)
constexpr size_t OFF_SBOX  = OFF_CAND  + (size_t)NN * 8;         // KPRE * float4
constexpr size_t OFF_SKEY  = OFF_SBOX  + (size_t)KPRE * 16;      // KPRE * u32
constexpr size_t OFF_MASK  = OFF_SKEY  + (size_t)KPRE * 4;       // KPRE * NH * u16

// ---------------- CDNA5 async global->LDS helpers ----------------
__device__ __forceinline__ unsigned lds_addr_of(const void* p) {
  // LDS generic VA: low 32 bits are the wave-relative LDS byte address.
  return (unsigned)(uintptr_t)p;
}
__device__ __forceinline__ void async_load_lds_b64(unsigned lds_addr, const void* gaddr) {
  asm volatile("global_load_async_to_lds_b64 %0, %1, off"
               :: "v"(lds_addr), "v"((unsigned long long)(uintptr_t)gaddr)
               : "memory");
}
__device__ __forceinline__ void async_load_lds_b128(unsigned lds_addr, const void* gaddr) {
  asm volatile("global_load_async_to_lds_b128 %0, %1, off"
               :: "v"(lds_addr), "v"((unsigned long long)(uintptr_t)gaddr)
               : "memory");
}
__device__ __forceinline__ void wait_async0() {
  asm volatile("s_wait_asynccnt 0" ::: "memory");
}

// order-preserving float -> u32 (total order, -inf smallest)
__device__ __forceinline__ unsigned f2key(float f) {
  unsigned u = __float_as_uint(f);
  return (u & 0x80000000u) ? ~u : (u | 0x80000000u);
}

// ---------------- kernel 0: zero hist + meta ----------------
__global__ void k_zero(unsigned* __restrict__ hist, unsigned* __restrict__ meta) {
  int t = blockIdx.x * blockDim.x + threadIdx.x;
  if (t < NBUCKET) hist[t] = 0u;
  if (t < 16)      meta[t] = 0u;
}

// ---------------- kernel 1: decode + score + key + histogram ----------------
__global__ void k_decode(const float* __restrict__ cls, const float* __restrict__ reg,
                         const float* __restrict__ anc,
                         float4* __restrict__ boxes, unsigned* __restrict__ keys,
                         unsigned* __restrict__ hist) {
  int n = blockIdx.x * blockDim.x + threadIdx.x;
  if (n >= NN) return;

  float l0 = cls[2 * n], l1 = cls[2 * n + 1];
  float prob = 1.0f / (1.0f + expf(l0 - l1));   // softmax prob of class 1

  float ax1 = anc[4*n], ay1 = anc[4*n+1], ax2 = anc[4*n+2], ay2 = anc[4*n+3];
  float acx = 0.5f * (ax1 + ax2), acy = 0.5f * (ay1 + ay2);
  float aw = ax2 - ax1, ah = ay2 - ay1;

  float rx = reg[4*n], ry = reg[4*n+1], rw = reg[4*n+2], rh = reg[4*n+3];
  float cx = rx * aw + acx, cy = ry * ah + acy;
  float bw = expf(rw) * aw, bh = expf(rh) * ah;

  float x1 = fminf(fmaxf(cx - 0.5f * bw, 0.f), 1.f);
  float y1 = fminf(fmaxf(cy - 0.5f * bh, 0.f), 1.f);
  float x2 = fminf(fmaxf(cx + 0.5f * bw, 0.f), 1.f);
  float y2 = fminf(fmaxf(cy + 0.5f * bh, 0.f), 1.f);

  bool ok = ((y2 - y1) >= MIN_SZ) && ((x2 - x1) >= MIN_SZ);
  float score = ok ? prob : -__builtin_inff();
  unsigned key = f2key(score);

  boxes[n] = make_float4(x1, y1, x2, y2);
  keys[n]  = key;
  atomicAdd(&hist[key >> 16], 1u);
}

// ---------------- kernel 2: suffix scan of histogram, pick threshold bucket ----------------
__global__ void k_scan(const unsigned* __restrict__ hist,
                       unsigned* __restrict__ cumge, unsigned* __restrict__ meta) {
  __shared__ unsigned sa[1024];
  __shared__ unsigned sb[1024];
  int t = threadIdx.x;                 // 0..1023, each owns 64 buckets
  int base = t * 64;

  unsigned sum = 0;
  for (int k = 0; k < 64; ++k) sum += hist[base + k];
  sa[t] = sum;
  __syncthreads();

  unsigned* cur = sa; unsigned* nxt = sb;
  for (int off = 1; off < 1024; off <<= 1) {          // inclusive suffix scan of chunk sums
    unsigned v = cur[t];
    if (t + off < 1024) v += cur[t + off];
    nxt[t] = v;
    __syncthreads();
    unsigned* tmp = cur; cur = nxt; nxt = tmp;
  }
  unsigned run = (t < 1023) ? cur[t + 1] : 0u;        // sum of all buckets above my chunk
  for (int k = 63; k >= 0; --k) {
    run += hist[base + k];
    cumge[base + k] = run;                            // #elements with bucket >= b
  }
  // B* = largest b with cumge[b] >= KPRE  (cumge is non-increasing)
  unsigned best = 0;
  for (int k = 63; k >= 0; --k) {
    if (cumge[base + k] >= (unsigned)KPRE) { best = (unsigned)(base + k); break; }
  }
  atomicMax(&meta[0], best);
}

// ---------------- kernel 3: compact candidates (bucket >= B*) ----------------
__global__ void k_select(const unsigned* __restrict__ keys, unsigned* __restrict__ meta,
                         uint2* __restrict__ cand) {
  int n = blockIdx.x * blockDim.x + threadIdx.x;
  if (n >= NN) return;
  unsigned bstar = meta[0];
  unsigned key = keys[n];
  if ((key >> 16) >= bstar) {
    unsigned pos = atomicAdd(&meta[1], 1u);
    cand[pos] = make_uint2(key, (unsigned)n);
  }
}

// ---------------- kernel 4: exact stable rank of each candidate ----------------
__global__ void k_rank(const uint2* __restrict__ cand, const unsigned* __restrict__ meta,
                       const float4* __restrict__ boxes,
                       float4* __restrict__ sbox, unsigned* __restrict__ skey) {
  __shared__ uint2 tile[256];
  unsigned C = meta[1];                               // >= KPRE by construction
  unsigned tid = blockIdx.x * 256u + threadIdx.x;
  uint2 me = cand[min(tid, C - 1u)];
  bool active = tid < C;

  unsigned r = 0;
  for (unsigned t0 = 0; t0 < C; t0 += 256u) {
    unsigned g = min(t0 + threadIdx.x, C - 1u);
    async_load_lds_b64(lds_addr_of(&tile[threadIdx.x]), &cand[g]);   // CDNA5 async -> LDS
    __builtin_prefetch(&cand[min(t0 + 256u + threadIdx.x, C - 1u)], 0, 0);
    wait_async0();
    __syncthreads();
    unsigned lim = min(256u, C - t0);
    for (unsigned j = 0; j < lim; ++j) {
      uint2 o = tile[j];
      r += (unsigned)((o.x > me.x) || (o.x == me.x && o.y < me.y));
    }
    __syncthreads();
  }
  if (active && r < (unsigned)KPRE) {
    sbox[r] = boxes[me.y];
    skey[r] = me.x;
  }
}

// ---------------- kernel 5: pairwise suppression bitmask (WMMA outer-sum) ----------------
// One wave32 per 16x16 (row,col) tile. Union term area_i + area_j is a rank-2
// outer product computed with V_WMMA_F32_16X16X4_F32:
//   A(16x4) = [area_row | 1 | 0 | 0],  B(4x16) = [1 ; area_col ; 0 ; 0]
//   => D[m][n] = area_row[m] + area_col[n]
// D layout: VGPR r, lane l  <->  (row m = r + 8*(l/16), col n = l%16).
// A layout: lanes 0-15 carry K0,K1 for M=lane; lanes 16-31 carry K2,K3 (zeros here).
// B layout (symmetric striping): lanes 0-15 carry K0,K1 for N=lane; lanes 16-31 K2,K3.
__global__ void __launch_bounds__(32)
k_mask(const float4* __restrict__ sbox, const unsigned* __restrict__ skey,
       unsigned short* __restrict__ mask16) {
  __shared__ float4  tb[32];      // [0..15] row boxes, [16..31] col boxes
  __shared__ unsigned srk[16];    // row keys (validity)
  __shared__ unsigned sball[8];   // per-r ballots

  int l  = threadIdx.x;
  int i0 = blockIdx.y * 16;       // row tile base
  int j0 = blockIdx.x * 16;       // col tile base
  int c  = l & 15;
  int h  = l >> 4;                // lane half

  // one async b128: lanes 0-15 stage the 16 row boxes, lanes 16-31 the 16 col boxes
  const float4* src = (h == 0) ? &sbox[i0 + c] : &sbox[j0 + c];
  async_load_lds_b128(lds_addr_of(&tb[l]), src);
  if (h == 0) srk[c] = skey[i0 + c];
  wait_async0();
  __syncthreads();

  float4 cbx = tb[16 + c];
  float areac = (cbx.z - cbx.x) * (cbx.w - cbx.y);
  float4 rbx  = tb[c];
  float arear = (rbx.z - rbx.x) * (rbx.w - rbx.y);

  v2f a, b;
  a.x = (h == 0) ? arear : 0.f;   // A[M=c][K0]   (K2 = 0)
  a.y = (h == 0) ? 1.f   : 0.f;   // A[M=c][K1]   (K3 = 0)
  b.x = (h == 0) ? 1.f   : 0.f;   // B[K0][N=c]   (K2 = 0)
  b.y = (h == 0) ? areac : 0.f;   // B[K1][N=c]   (K3 = 0)
  v8f acc = {};
  acc = __builtin_amdgcn_wmma_f32_16x16x4_f32(false, a, false, b,
                                              (short)0, acc, false, false);

  #pragma unroll
  for (int r = 0; r < 8; ++r) {
    int m = r + 8 * h;                       // my row for this D register
    float4 rb = tb[m];
    bool vr = srk[m] >= 0x80000000u;         // row passed size filter
    float lx = fmaxf(rb.x, cbx.x), ly = fmaxf(rb.y, cbx.y);
    float rx = fminf(rb.z, cbx.z), ry = fminf(rb.w, cbx.w);
    float ww = fmaxf(rx - lx, 0.f), hh = fmaxf(ry - ly, 0.f);
    float inter = ww * hh;
    float uni = fmaxf(acc[r] - inter, 1e-12f);   // area_m + area_n - inter (WMMA)
    int gi = i0 + m, gj = j0 + c;
    bool sup = vr && (gj > gi) && (inter > IOUTHR * uni);
    unsigned bv = __builtin_amdgcn_ballot_w32(sup);
    // ballot bit l' = (row r + 8*(l'/16), col l'%16): halves are two row-masks
    if (l == 0) sball[r] = bv;
  }
  __syncthreads();
  if (l < 16) {
    unsigned short hw = (l < 8) ? (unsigned short)(sball[l] & 0xFFFFu)
                                : (unsigned short)(sball[l - 8] >> 16);
    mask16[(size_t)(i0 + l) * NH + blockIdx.x] = hw;
  }
}

// ---------------- kernel 6: sequential greedy scan + compaction ----------------
__global__ void __launch_bounds__(256)
k_nms_scan(const unsigned short* __restrict__ mask16, const unsigned* __restrict__ skey,
           const float4* __restrict__ sbox, float* __restrict__ out) {
  __shared__ unsigned rem[NWORDS];
  __shared__ unsigned validW[NWORDS];
  __shared__ unsigned keptW[NWORDS];
  __shared__ unsigned pa[256];
  __shared__ unsigned pb[256];
  int t = threadIdx.x;

  if (t < NWORDS) {
    rem[t] = 0u;
    unsigned vw = 0u;
    for (int b = 0; b < 32; ++b) {
      int i = t * 32 + b;
      if (i < KPRE && skey[i] >= 0x80000000u) vw |= 1u << b;
    }
    validW[t] = vw;
  }
  unsigned myKept = 0u;
  __syncthreads();

  for (int i = 0; i < KPRE; ++i) {
    int w = i >> 5, b = i & 31;
    bool sup   = (rem[w] >> b) & 1u;
    bool valid = (validW[w] >> b) & 1u;
    bool keep  = valid && !sup;                 // uniform across the block
    __syncthreads();                            // all reads of rem[] done
    if (keep && t < NWORDS) {
      const unsigned short* mrow = &mask16[(size_t)i * NH];
      unsigned lo = mrow[2 * t];
      unsigned hi = (2 * t + 1 < NH) ? (unsigned)mrow[2 * t + 1] : 0u;
      rem[t] |= lo | (hi << 16);
    }
    if (t < NWORDS && i + 1 < KPRE)
      __builtin_prefetch(&mask16[(size_t)(i + 1) * NH + 2 * t], 0, 0);
    if (t == w) myKept |= ((unsigned)keep) << b;
    __syncthreads();                            // rem[] updates visible
  }

  if (t < NWORDS) keptW[t] = myKept;
  pa[t] = (t < NWORDS) ? (unsigned)__popc(myKept) : 0u;
  __syncthreads();
  unsigned* cur = pa; unsigned* nxt = pb;
  for (int off = 1; off < 256; off <<= 1) {     // inclusive prefix of word popcounts
    unsigned v = cur[t];
    if (t >= off) v += cur[t - off];
    nxt[t] = v;
    __syncthreads();
    unsigned* tmp = cur; cur = nxt; nxt = tmp;
  }
  unsigned total = cur[255];                    // total kept count

  for (int i = t; i < KPRE; i += 256) {
    int w = i >> 5, b = i & 31;
    unsigned kw = keptW[w];
    if ((kw >> b) & 1u) {
      unsigned rank = (w > 0 ? cur[w - 1] : 0u) + (unsigned)__popc(kw & ((1u << b) - 1u));
      if (rank < (unsigned)KPOST) {
        float4 bx = sbox[i];
        out[rank * 4 + 0] = bx.x; out[rank * 4 + 1] = bx.y;
        out[rank * 4 + 2] = bx.z; out[rank * 4 + 3] = bx.w;
      }
    }
  }
  // zero only the unwritten tail (disjoint from kept slots -> no ordering hazard)
  for (int s = t; s < KPOST; s += 256) {
    if ((unsigned)s >= total) {
      out[s * 4 + 0] = 0.f; out[s * 4 + 1] = 0.f;
      out[s * 4 + 2] = 0.f; out[s * 4 + 3] = 0.f;
    }
  }
}

// ---------------- host launcher ----------------
extern "C" void kernel_launch(void* const* d_in, const int* in_sizes, int n_in,
                              void* d_out, int out_size, void* d_ws, size_t ws_size,
                              hipStream_t stream) {
  const float* cls = (const float*)d_in[0];   // (1,100,100,18)
  const float* reg = (const float*)d_in[1];   // (1,90000,4)
  const float* anc = (const float*)d_in[2];   // (90000,4)
  float* out = (float*)d_out;                 // (300,4)

  char* ws = (char*)d_ws;
  float4*         boxes  = (float4*)        (ws + OFF_BOXES);
  unsigned*       keys   = (unsigned*)      (ws + OFF_KEYS);
  unsigned*       hist   = (unsigned*)      (ws + OFF_HIST);
  unsigned*       cum    = (unsigned*)      (ws + OFF_CUM);
  unsigned*       meta   = (unsigned*)      (ws + OFF_META);
  uint2*          cand   = (uint2*)         (ws + OFF_CAND);
  float4*         sbox   = (float4*)        (ws + OFF_SBOX);
  unsigned*       skey   = (unsigned*)      (ws + OFF_SKEY);
  unsigned short* mask16 = (unsigned short*)(ws + OFF_MASK);
  (void)cum;

  const int TB = 256;
  k_zero  <<<(NBUCKET + TB - 1) / TB, TB, 0, stream>>>(hist, meta);
  k_decode<<<(NN + TB - 1) / TB,      TB, 0, stream>>>(cls, reg, anc, boxes, keys, hist);
  k_scan  <<<1, 1024,                     0, stream>>>(hist, cum, meta);
  k_select<<<(NN + TB - 1) / TB,      TB, 0, stream>>>(keys, meta, cand);
  k_rank  <<<(NN + TB - 1) / TB,      TB, 0, stream>>>(cand, meta, boxes, sbox, skey);
  k_mask  <<<dim3(NH, NTILE), dim3(32),   0, stream>>>(sbox, skey, mask16);
  k_nms_scan<<<1, 256,                    0, stream>>>(mask16, skey, sbox, out);
}